// SigMMD_63823214018691
// MI455X (gfx1250) — compile-verified
//
#include <hip/hip_runtime.h>

// ---------------------------------------------------------------------------
// Signature-kernel MMD, fused WMMA(f32) GEMM + Goursat PDE sweep, gfx1250.
//   X, Y : (64, 128, 64) f32.  One wave32 per (gram, a, b) task (3*4096 tasks).
//   Per task: dX^T, dY^T staged in LDS (stride-129 padded, conflict-free),
//   inc computed 16 rows at a time with V_WMMA_F32_16X16X4_F32, PDE row
//   recurrence done with in-lane prefix + Kogge-Stone wave scan.
//   v2: A-matrix operands hoisted out of the j-tile loop.
//   v3: exploit Gram symmetry for XX/YY (compute upper triangle only, mirror
//       the result) -> ~33% less total WMMA/PDE work; all ws entries still
//       written exactly once (deterministic, no atomics).
// ---------------------------------------------------------------------------

typedef __attribute__((ext_vector_type(2))) float v2f;
typedef __attribute__((ext_vector_type(8))) float v8f;

#define TLEN     128
#define DIM      64
#define M1       127            // TLEN - 1
#define TSTRIDE  129            // padded LDS row stride (bank-conflict-free)
#define LDS_XT   (DIM * TSTRIDE)          // 8256 floats per transposed diff
#define LDS_FLOATS (2 * LDS_XT + 16 * 128)

__global__ __launch_bounds__(32)
void sig_gram_pde(const float* __restrict__ X,
                  const float* __restrict__ Y,
                  float* __restrict__ ws) {
  extern __shared__ float smem[];
  float* sXT  = smem;                 // [DIM][TSTRIDE]  dX transposed (k-major)
  float* sYT  = smem + LDS_XT;        // [DIM][TSTRIDE]  dY transposed (k-major)
  float* sInc = smem + 2 * LDS_XT;    // [16][128] one inc row-block

  const int lane = threadIdx.x;       // wave32
  const unsigned bid = blockIdx.x;    // 0 .. 3*4096-1
  const int g = bid >> 12;            // gram: 0=XX 1=YY 2=XY
  const int p = bid & 4095;
  const int a = p >> 6;
  const int b = p & 63;

  // Symmetric grams: only compute the upper triangle (a <= b); the (a,b)
  // task also writes the mirrored (b,a) entry below.
  if (g < 2 && a > b) return;

  const float* P = (g == 1) ? Y : X;
  const float* Q = (g == 0) ? X : Y;
  const float  w = (g == 2) ? (-2.0f / 4096.0f) : (1.0f / 4096.0f);

  const float* Pr = P + a * (TLEN * DIM);
  const float* Qr = Q + b * (TLEN * DIM);

  // ---- stage path increments, transposed, into LDS (coalesced global reads,
  //      stride-129 LDS writes => distinct banks across lanes) ----
  for (int idx = lane; idx < M1 * DIM; idx += 32) {
    int r = idx >> 6, k = idx & 63;
    sXT[k * TSTRIDE + r] = Pr[idx + DIM] - Pr[idx];
    sYT[k * TSTRIDE + r] = Qr[idx + DIM] - Qr[idx];
  }
  for (int k = lane; k < DIM; k += 32) {   // zero-pad column 127 (i=127/j=127)
    sXT[k * TSTRIDE + M1] = 0.0f;
    sYT[k * TSTRIDE + M1] = 0.0f;
  }

  const int half = lane >> 4;   // 0: lanes 0-15, 1: lanes 16-31
  const int q    = lane & 15;

  // PDE state: K row (128 wide), 4 values per lane, j = 4*lane + m
  float kp0 = 1.0f, kp1 = 1.0f, kp2 = 1.0f, kp3 = 1.0f;

  for (int iblk = 0; iblk < M1; iblk += 16) {
    // ---- A operands (16x64 dX block) loaded once, reused for all 8 tiles ---
    v2f areg[16];
#pragma unroll
    for (int kk = 0; kk < 16; ++kk) {
      const int k0 = kk * 4;
      // A (16x4, MxK): lane row M=q; VGPR0/1 hold K = k0+2*half (+1)
      areg[kk].x = sXT[(k0 + 2 * half    ) * TSTRIDE + iblk + q];
      areg[kk].y = sXT[(k0 + 2 * half + 1) * TSTRIDE + iblk + q];
    }

    // ---- inc[iblk..iblk+15][0..127] via 8 tiles x 16 f32 WMMAs (K=64) ----
    for (int t = 0; t < 8; ++t) {
      v8f acc = {0.f, 0.f, 0.f, 0.f, 0.f, 0.f, 0.f, 0.f};
#pragma unroll
      for (int kk = 0; kk < 16; ++kk) {
        const int k0 = kk * 4;
        v2f bv;
        // B (4x16, KxN): lane col N=q; VGPR0/1 hold K = k0+half (+2)
        bv.x = sYT[(k0 + half      ) * TSTRIDE + t * 16 + q];
        bv.y = sYT[(k0 + half + 2  ) * TSTRIDE + t * 16 + q];
        acc = __builtin_amdgcn_wmma_f32_16x16x4_f32(
            false, areg[kk], false, bv, (short)0, acc, false, false);
      }
      // C layout: VGPR r -> row M = r + 8*half, col N = q
#pragma unroll
      for (int r = 0; r < 8; ++r)
        sInc[(r + 8 * half) * 128 + t * 16 + q] = acc[r];
    }

    // ---- PDE row sweep over this block (sequential in i) ----
    const int imax = (M1 - iblk < 16) ? (M1 - iblk) : 16;
    for (int ii = 0; ii < imax; ++ii) {
      const float4 iv =
          *reinterpret_cast<const float4*>(&sInc[ii * 128 + lane * 4]);
      float kn = __shfl_down(kp0, 1, 32);      // kprev[4*lane+4]
      if (lane == 31) kn = kp3;                // makes delta[127] exactly 0
      float d0 = kp1 - kp0 + kp0 * iv.x;
      float d1 = kp2 - kp1 + kp1 * iv.y;
      float d2 = kp3 - kp2 + kp2 * iv.z;
      float d3 = kn  - kp3 + kp3 * iv.w;
      float s0 = d0, s1 = s0 + d1, s2 = s1 + d2, s3 = s2 + d3;
      // Kogge-Stone inclusive scan of per-lane totals
      float sc = s3;
#pragma unroll
      for (int off = 1; off < 32; off <<= 1) {
        float y = __shfl_up(sc, off, 32);
        if (lane >= off) sc += y;
      }
      const float ex = sc - s3;                // exclusive prefix of lane sums
      kp0 = 1.0f + ex;                         // k_new[4l+0]
      kp1 = 1.0f + ex + s0;
      kp2 = 1.0f + ex + s1;
      kp3 = 1.0f + ex + s2;                    // k_new[4l+3]
    }
  }

  // K(a,b) = k[127] lives in lane 31, m=3
  if (lane == 31) {
    const float val = w * kp3;
    ws[(g << 12) + (a << 6) + b] = val;
    if (g < 2 && a < b) ws[(g << 12) + (b << 6) + a] = val;  // mirror
  }
}

// ---------------------------------------------------------------------------
// Deterministic final reduction: sum weighted gram entries + boundary MSE term
// ---------------------------------------------------------------------------
__global__ __launch_bounds__(256)
void sig_reduce(const float* __restrict__ X, const float* __restrict__ Y,
                const float* __restrict__ ws, float* __restrict__ out) {
  __shared__ float red[256];
  const int tid = threadIdx.x;
  float acc = 0.0f;
  for (int i = tid; i < 3 * 4096; i += 256) acc += ws[i];
  for (int i = tid; i < 4096; i += 256) {
    const int a = i >> 6, k = i & 63;
    const float d = X[a * (TLEN * DIM) + k] - Y[a * (TLEN * DIM) + k];
    acc += d * d * (1.0f / 4096.0f);
  }
  red[tid] = acc;
  __syncthreads();
  for (int s = 128; s > 0; s >>= 1) {
    if (tid < s) red[tid] += red[tid + s];
    __syncthreads();
  }
  if (tid == 0) out[0] = red[0];
}

extern "C" void kernel_launch(void* const* d_in, const int* in_sizes, int n_in,
                              void* d_out, int out_size, void* d_ws, size_t ws_size,
                              hipStream_t stream) {
  const float* X = (const float*)d_in[0];
  const float* Y = (const float*)d_in[1];
  float* ws  = (float*)d_ws;
  float* out = (float*)d_out;

  const size_t lds_bytes = LDS_FLOATS * sizeof(float);   // ~72.5 KB per WG
  sig_gram_pde<<<dim3(3 * 4096), dim3(32), lds_bytes, stream>>>(X, Y, ws);
  sig_reduce<<<dim3(1), dim3(256), 0, stream>>>(X, Y, ws, out);
}